// GCN_13443247637120
// MI455X (gfx1250) — compile-verified
//
#include <hip/hip_runtime.h>
#include <hip/hip_bf16.h>

typedef __attribute__((ext_vector_type(2))) float v2f;
typedef __attribute__((ext_vector_type(8))) float v8f;

#define IN_SIZE 128
#define OUT_SIZE 64
#define WP_STRIDE 160   // K-pair row stride in floats; 160 % 64 == 32 banks ->
                        // halves 0/1 hit disjoint LDS bank sets (conflict-free)

// ---------------------------------------------------------------------------
// Kernel 1: support = x @ weight  (fp32 WMMA 16x16x4)
// One wave computes a 16(M) x 64(N) tile. 8 waves per block.
// Weight staged in LDS pre-swizzled into K-pairs: element (k, col) lives at
// wlds[(k>>1)*WP_STRIDE + col*2 + (k&1)], so every B fragment is one aligned
// 8-byte LDS load (no v_mov shuffle chains feeding the WMMAs).
// ---------------------------------------------------------------------------
__global__ __launch_bounds__(256) void gcn_gemm_wmma(
    const float* __restrict__ x,       // [N, 128] row-major
    const float* __restrict__ weight,  // [128, 64] row-major
    float* __restrict__ support,       // [N, 64]
    int n_nodes, int n_tiles)
{
    __shared__ float wlds[(IN_SIZE / 2) * WP_STRIDE];  // 40 KB

    // cooperative swizzled load of weight into LDS
    for (int i = threadIdx.x; i < IN_SIZE * OUT_SIZE; i += 256) {
        const int k   = i >> 6;    // 0..127
        const int col = i & 63;    // 0..63
        wlds[(k >> 1) * WP_STRIDE + col * 2 + (k & 1)] = weight[i];
    }
    __syncthreads();

    const int wave = threadIdx.x >> 5;         // 0..7
    const int lane = threadIdx.x & 31;
    const int tileM = blockIdx.x * 8 + wave;
    if (tileM >= n_tiles) return;

    const int half = lane >> 4;                // 0: K=k,k+1 ; 1: K=k+2,k+3
    const int l16  = lane & 15;

    int row = tileM * 16 + l16;                // A-matrix M for this lane
    if (row >= n_nodes) row = n_nodes - 1;     // clamp (N is a multiple of 16)
    const float* xrow = x + (size_t)row * IN_SIZE;

    v8f c0 = {}, c1 = {}, c2 = {}, c3 = {};

    #pragma unroll 4
    for (int kk = 0; kk < IN_SIZE / 4; ++kk) {
        const int k4 = kk * 4;

        // A fragment: 16x4 f32 layout -> lane holds (M=l16, K=k4+2*half, +1)
        v2f a = *(const v2f*)(xrow + k4 + 2 * half);

        // B fragments: K-pair row (kk*2 + half), columns nt*16 + l16
        const float* brow = wlds + (kk * 2 + half) * WP_STRIDE + l16 * 2;
        v2f b0 = *(const v2f*)(brow + 0);
        v2f b1 = *(const v2f*)(brow + 32);
        v2f b2 = *(const v2f*)(brow + 64);
        v2f b3 = *(const v2f*)(brow + 96);

        c0 = __builtin_amdgcn_wmma_f32_16x16x4_f32(false, a, false, b0, (short)0, c0, false, false);
        c1 = __builtin_amdgcn_wmma_f32_16x16x4_f32(false, a, false, b1, (short)0, c1, false, false);
        c2 = __builtin_amdgcn_wmma_f32_16x16x4_f32(false, a, false, b2, (short)0, c2, false, false);
        c3 = __builtin_amdgcn_wmma_f32_16x16x4_f32(false, a, false, b3, (short)0, c3, false, false);
    }

    // Store D: VGPR j holds M = j + 8*half, N = l16 (+16*nt)
    #pragma unroll
    for (int j = 0; j < 8; ++j) {
        const int orow = tileM * 16 + j + 8 * half;
        if (orow < n_nodes) {
            float* dst = support + (size_t)orow * OUT_SIZE + l16;
            dst[0]  = c0[j];
            dst[16] = c1[j];
            dst[32] = c2[j];
            dst[48] = c3[j];
        }
    }
}

// ---------------------------------------------------------------------------
// Kernel 2: out[i, :] = bias[:]
// ---------------------------------------------------------------------------
__global__ void bias_init(float* __restrict__ out,
                          const float* __restrict__ bias, int total)
{
    int i = blockIdx.x * blockDim.x + threadIdx.x;
    if (i < total) out[i] = bias[i & (OUT_SIZE - 1)];
}

// ---------------------------------------------------------------------------
// Kernel 3: for each edge e: out[rows[e], :] += support[cols[e], :] * values[e]
// 16 lanes per edge, float4 per lane, hardware f32 atomics (L2-resident).
// ---------------------------------------------------------------------------
__global__ void edge_scatter(const int* __restrict__ rows,
                             const int* __restrict__ cols,
                             const float* __restrict__ values,
                             const float* __restrict__ support,
                             float* __restrict__ out, int n_edges)
{
    int t = blockIdx.x * blockDim.x + threadIdx.x;
    int e = t >> 4;
    if (e >= n_edges) return;
    int lane4 = (t & 15) * 4;

    int r = rows[e];
    int c = cols[e];
    float v = values[e];

    float4 s = *(const float4*)(support + (size_t)c * OUT_SIZE + lane4);
    float* dst = out + (size_t)r * OUT_SIZE + lane4;

    unsafeAtomicAdd(dst + 0, s.x * v);
    unsafeAtomicAdd(dst + 1, s.y * v);
    unsafeAtomicAdd(dst + 2, s.z * v);
    unsafeAtomicAdd(dst + 3, s.w * v);
}

// ---------------------------------------------------------------------------
extern "C" void kernel_launch(void* const* d_in, const int* in_sizes, int n_in,
                              void* d_out, int out_size, void* d_ws, size_t ws_size,
                              hipStream_t stream)
{
    const float* x      = (const float*)d_in[0];
    const int*   rows   = (const int*)d_in[1];
    const int*   cols   = (const int*)d_in[2];
    const float* values = (const float*)d_in[3];
    const float* weight = (const float*)d_in[4];
    const float* bias   = (const float*)d_in[5];
    float* out = (float*)d_out;

    const int n_nodes = in_sizes[0] / IN_SIZE;   // 50000
    const int n_edges = in_sizes[1];             // 800000

    float* support = (float*)d_ws;               // [n_nodes, 64] = 12.8 MB

    // 1) dense transform with fp32 WMMA
    const int n_tiles = (n_nodes + 15) / 16;     // 3125
    const int gemm_blocks = (n_tiles + 7) / 8;
    gcn_gemm_wmma<<<gemm_blocks, 256, 0, stream>>>(x, weight, support,
                                                   n_nodes, n_tiles);

    // 2) out = bias (broadcast)
    const int total = n_nodes * OUT_SIZE;
    bias_init<<<(total + 255) / 256, 256, 0, stream>>>(out, bias, total);

    // 3) scatter-accumulate edges
    const long long ethreads = (long long)n_edges * 16;
    const int eblocks = (int)((ethreads + 255) / 256);
    edge_scatter<<<eblocks, 256, 0, stream>>>(rows, cols, values, support,
                                              out, n_edges);
}